// MultiHeadAttentionQuantum_65481071402693
// MI455X (gfx1250) — compile-verified
//
#include <hip/hip_runtime.h>
#include <math.h>

#define NB    8
#define SEQ   2048
#define NE    8
#define NW    4          // kernel wires
#define NF    16         // 2^NW features
#define TILES (SEQ / 16) // 128 column tiles
#define WPB   4          // waves per block

typedef float v2f __attribute__((ext_vector_type(2)));
typedef float v8f __attribute__((ext_vector_type(8)));

// ---------------------------------------------------------------------------
// Kernel 1: per-token quantum-layer collapse + half-angle feature expansion.
//   q_w  = prod cos(x+phi) over the CNOT-ring XOR subsets (prefix products)
//   Qfeat[m] = prod_w (m_w ? sin(q_w/2) : cos(q_w/2))     (16 features)
//   Kfeat stored transposed [B, 16, S] for coalesced WMMA B-operand loads.
// ---------------------------------------------------------------------------
__global__ __launch_bounds__(256)
void qfeat_kernel(const float* __restrict__ x,
                  const float* __restrict__ phi_q,
                  const float* __restrict__ phi_k,
                  const float* __restrict__ phi_v,
                  float* __restrict__ Qf,
                  float* __restrict__ KT,
                  float* __restrict__ V)
{
    int t = blockIdx.x * blockDim.x + threadIdx.x;
    if (t >= NB * SEQ) return;
    int bi = t / SEQ;
    int si = t % SEQ;

    float xv[NE];
#pragma unroll
    for (int i = 0; i < NE; i++) xv[i] = x[t * NE + i];

    // ---- q layer (only wires 0..3 feed the kernel similarity) ----
    float zq[NW];
    {
        float c[NE];
#pragma unroll
        for (int i = 0; i < NE; i++) c[i] = __cosf(xv[i] + phi_q[i]);
        float suf = 1.0f;
#pragma unroll
        for (int i = 1; i < NE; i++) suf *= c[i];
        zq[0] = suf;                       // wire 0: XOR of bits 1..7
        float run = c[0];
#pragma unroll
        for (int w = 1; w < NW; w++) { run *= c[w]; zq[w] = run; }
    }
    // ---- k layer ----
    float zk[NW];
    {
        float c[NE];
#pragma unroll
        for (int i = 0; i < NE; i++) c[i] = __cosf(xv[i] + phi_k[i]);
        float suf = 1.0f;
#pragma unroll
        for (int i = 1; i < NE; i++) suf *= c[i];
        zk[0] = suf;
        float run = c[0];
#pragma unroll
        for (int w = 1; w < NW; w++) { run *= c[w]; zk[w] = run; }
    }
    // ---- v layer (all 8 wires) ----
    {
        float c[NE], vv[NE];
#pragma unroll
        for (int i = 0; i < NE; i++) c[i] = __cosf(xv[i] + phi_v[i]);
        float suf = 1.0f;
#pragma unroll
        for (int i = 1; i < NE; i++) suf *= c[i];
        vv[0] = suf;
        float run = c[0];
#pragma unroll
        for (int w = 1; w < NE; w++) { run *= c[w]; vv[w] = run; }
#pragma unroll
        for (int w = 0; w < NE; w++) V[(size_t)t * NE + w] = vv[w];
    }

    // ---- half-angle feature expansion: cos(0.5(q-k)) pair expansion ----
    float cq[NW], sq[NW], ck[NW], sk[NW];
#pragma unroll
    for (int w = 0; w < NW; w++) {
        cq[w] = __cosf(0.5f * zq[w]); sq[w] = __sinf(0.5f * zq[w]);
        ck[w] = __cosf(0.5f * zk[w]); sk[w] = __sinf(0.5f * zk[w]);
    }
#pragma unroll
    for (int m = 0; m < NF; m++) {
        float pq = 1.0f, pk = 1.0f;
#pragma unroll
        for (int w = 0; w < NW; w++) {
            pq *= (m >> w & 1) ? sq[w] : cq[w];
            pk *= (m >> w & 1) ? sk[w] : ck[w];
        }
        Qf[(size_t)t * NF + m]                    = pq;
        KT[((size_t)bi * NF + m) * SEQ + si]      = pk;
    }
}

// ---------------------------------------------------------------------------
// Kernel 2: fused attention. One wave32 owns a 16-row tile of one batch.
//   scores tile (16x16) = Qfeat(16x16) x KfeatT(16x16) via 4x wmma f32 16x16x4
//   P = exp(|scores|)  (sim in [0,1] -> softmax needs no max subtraction)
//   acc += P x V_ext, where V_ext col 8 == 1.0 -> acc col 8 = softmax denom.
//   Finish: divide, then tiny 8x8 linear (out = attn @ W^T + b) per row.
// ---------------------------------------------------------------------------
__global__ __launch_bounds__(32 * WPB)
void qattn_kernel(const float* __restrict__ Qf,
                  const float* __restrict__ KT,
                  const float* __restrict__ V,
                  const float* __restrict__ W,
                  const float* __restrict__ bias,
                  float* __restrict__ out)
{
    __shared__ float lds[WPB][16][17];   // per-wave 16x16 staging (+1 pad)

    const int lane = threadIdx.x & 31;
    const int wv   = threadIdx.x >> 5;
    const int g    = blockIdx.x * WPB + wv;       // global row-tile id
    const int bi   = g / TILES;
    const int tile = g % TILES;
    const int rowbase = tile * 16;

    const int half = lane >> 4;   // lane group: 0 -> K/M even pair, 1 -> +2
    const int nidx = lane & 15;   // column / row-within-group index

    // A operand: Qfeat rows (loop invariant). A-layout for f32 16x16x4:
    // lane L holds row (L%16); VGPR0 K = k0, VGPR1 K = k0+1, k0 = 4*cc + 2*(L/16).
    v2f aq[4];
    {
        const float* qrow = Qf + (size_t)(bi * SEQ + rowbase + nidx) * NF;
#pragma unroll
        for (int cc = 0; cc < 4; cc++) {
            int k0 = 4 * cc + 2 * half;
            aq[cc][0] = qrow[k0];
            aq[cc][1] = qrow[k0 + 1];
        }
    }

    const float* ktb = KT + (size_t)bi * NF * SEQ;
    const float* vb  = V  + (size_t)bi * SEQ * NE;

    v8f acc = {0.f, 0.f, 0.f, 0.f, 0.f, 0.f, 0.f, 0.f};

    for (int jt = 0; jt < TILES; jt++) {
        const int jg = jt * 16 + nidx;

        // ---- scores = Qfeat x KfeatT  (K = 16 via 4 chained K=4 WMMAs) ----
        v8f d = {0.f, 0.f, 0.f, 0.f, 0.f, 0.f, 0.f, 0.f};
#pragma unroll
        for (int cc = 0; cc < 4; cc++) {
            int k0 = 4 * cc + 2 * half;
            v2f bk;
            bk[0] = ktb[(size_t)k0 * SEQ + jg];        // coalesced: 16 lanes, 64B
            bk[1] = ktb[(size_t)(k0 + 1) * SEQ + jg];
            d = __builtin_amdgcn_wmma_f32_16x16x4_f32(
                    false, aq[cc], false, bk, (short)0, d, false, false);
        }

        // ---- P = exp(|sim|), stage C-layout -> LDS for A-layout reload ----
#pragma unroll
        for (int r = 0; r < 8; r++)
            lds[wv][r + 8 * half][nidx] = __expf(fabsf(d[r]));
        asm volatile("s_wait_dscnt 0" ::: "memory");   // RAW across lanes, same wave

        // ---- acc += P x V_ext (col 8 of V_ext = 1.0 -> row sums) ----
#pragma unroll
        for (int cc = 0; cc < 4; cc++) {
            int j0  = 4 * cc + 2 * half;
            int jg0 = jt * 16 + j0;
            v2f pa, bv;
            pa[0] = lds[wv][nidx][j0];
            pa[1] = lds[wv][nidx][j0 + 1];
            bv[0] = (nidx < 8) ? vb[(size_t)jg0 * NE + nidx]
                               : (nidx == 8 ? 1.0f : 0.0f);
            bv[1] = (nidx < 8) ? vb[(size_t)(jg0 + 1) * NE + nidx]
                               : (nidx == 8 ? 1.0f : 0.0f);
            acc = __builtin_amdgcn_wmma_f32_16x16x4_f32(
                      false, pa, false, bv, (short)0, acc, false, false);
        }
        // next-iter LDS stores are WAR vs these loads: DS is in-order per wave.
    }

    // ---- softmax normalize: denom lives in column N==8 (lanes 8 / 24) ----
#pragma unroll
    for (int r = 0; r < 8; r++) {
        float den = __shfl(acc[r], (lane & 16) + 8, 32);
        float a   = acc[r] / den;
        if (nidx < 8) lds[wv][r + 8 * half][nidx] = a;   // attn tile 16x8 -> LDS
    }
    asm volatile("s_wait_dscnt 0" ::: "memory");

    // ---- final linear: out = attn @ W^T + b  (E = 8) ----
    if (lane < 16) {
        const int row = lane;
        const size_t tok = (size_t)bi * SEQ + rowbase + row;
#pragma unroll
        for (int e = 0; e < NE; e++) {
            float s = bias[e];
#pragma unroll
            for (int n = 0; n < NE; n++)
                s += lds[wv][row][n] * W[e * NE + n];
            out[tok * NE + e] = s;
        }
    }
}

extern "C" void kernel_launch(void* const* d_in, const int* in_sizes, int n_in,
                              void* d_out, int out_size, void* d_ws, size_t ws_size,
                              hipStream_t stream)
{
    const float* x     = (const float*)d_in[0];
    const float* phi_q = (const float*)d_in[1];
    const float* phi_k = (const float*)d_in[2];
    const float* phi_v = (const float*)d_in[3];
    const float* W     = (const float*)d_in[4];
    const float* bias  = (const float*)d_in[5];
    float* out = (float*)d_out;

    // workspace layout (floats): Qfeat [B*S,16] | KfeatT [B,16,S] | V [B*S,8]
    float* Qf = (float*)d_ws;
    float* KT = Qf + (size_t)NB * SEQ * NF;
    float* V  = KT + (size_t)NB * NF * SEQ;

    qfeat_kernel<<<(NB * SEQ) / 256, 256, 0, stream>>>(x, phi_q, phi_k, phi_v,
                                                       Qf, KT, V);
    qattn_kernel<<<(NB * TILES) / WPB, 32 * WPB, 0, stream>>>(Qf, KT, V,
                                                              W, bias, out);
}